// DiscreteTemporalEmbedding_33981781246639
// MI455X (gfx1250) — compile-verified
//
#include <hip/hip_runtime.h>
#include <stdint.h>

// DiscreteTemporalEmbedding: out[b,n,c,t] = W[dow(b,t),c] + W[7+tod(b,t),c],
// broadcast over n. 402.7 MB of stores => pure HBM-store-bound (~17.3 us floor
// at 23.3 TB/s). Strategy: build the unique 6 KB [c][t] tile per batch in LDS,
// then stream it to HBM with CDNA5 async LDS->global b128 stores (ASYNCcnt),
// marked non-temporal so the write-once output doesn't churn the 192 MB L2.

#define B_   64
#define T_   24
#define N_   1024
#define C_   64
#define DPW_ 7
#define TPD_ 288

#define NCHUNK      16                 // vertices replicated per block
#define THREADS     384                // 12 waves; 384*16B == one full tile
#define TILE_FLOATS (C_ * T_)          // 1536 floats
#define TILE_BYTES  (TILE_FLOATS * 4)  // 6144 bytes

__global__ __launch_bounds__(THREADS)
void dte_broadcast_kernel(const int* __restrict__ tin,   // [B, T, 2] int32
                          const float* __restrict__ W,   // [DPW+TPD, C] f32
                          float* __restrict__ out)       // [B, N, C, T] f32
{
    __shared__ float tile[TILE_FLOATS];  // tile[c*T_ + t] == out inner block

    const int tid   = threadIdx.x;
    const int b     = blockIdx.x >> 6;   // 64 chunks per batch
    const int chunk = blockIdx.x & 63;

    // ---- Phase 1: compute the unique [c][t] tile (exact f32 gather+add) ----
#pragma unroll
    for (int k = 0; k < TILE_FLOATS / THREADS; ++k) {
        int idx = tid + k * THREADS;         // 0..1535
        int tt  = idx % T_;
        int c   = idx / T_;
        int dow = tin[(b * T_ + tt) * 2 + 0] % DPW_;
        int tod = tin[(b * T_ + tt) * 2 + 1] % TPD_;
        tile[idx] = W[dow * C_ + c] + W[(DPW_ + tod) * C_ + c];
    }
    __syncthreads();

    // ---- Phase 2: replicate tile over NCHUNK vertices via async stores ----
    // GVS mode: saddr = 64-bit SGPR base, vaddr = 32-bit per-lane byte offset,
    // vdata = VGPR holding the LDS byte address of this lane's 16-byte slice.
    // th:TH_STORE_NT: write-once streaming output, don't retain in caches.
    unsigned lds_off = ((unsigned)(uintptr_t)tile) + (unsigned)tid * 16u;
    unsigned g_off   = (unsigned)tid * 16u;
    uint64_t base    = (uint64_t)(uintptr_t)out +
                       ((uint64_t)(b * N_ + chunk * NCHUNK)) * (uint64_t)TILE_BYTES;

#pragma unroll
    for (int n = 0; n < NCHUNK; ++n) {
        asm volatile("global_store_async_from_lds_b128 %0, %1, %2 th:TH_STORE_NT"
                     :
                     : "v"(g_off), "v"(lds_off), "s"(base)
                     : "memory");
        g_off += (unsigned)TILE_BYTES;
    }

    // Drain ASYNCcnt before the LDS tile can be considered dead.
    asm volatile("s_wait_asynccnt 0" ::: "memory");
}

extern "C" void kernel_launch(void* const* d_in, const int* in_sizes, int n_in,
                              void* d_out, int out_size, void* d_ws, size_t ws_size,
                              hipStream_t stream) {
    const int*   tin = (const int*)d_in[0];    // t: [B, T, 2] int32
    const float* W   = (const float*)d_in[1];  // W: [295, 64] f32
    float*       out = (float*)d_out;          // [B, N, C, T] f32

    dim3 grid(B_ * (N_ / NCHUNK));  // 4096 blocks
    dim3 block(THREADS);            // 384 threads (12 waves)
    hipLaunchKernelGGL(dte_broadcast_kernel, grid, block, 0, stream, tin, W, out);
}